// Decode_82085414961544
// MI455X (gfx1250) — compile-verified
//
#include <hip/hip_runtime.h>
#include <hip/hip_bf16.h>
#include <math.h>

#define BB   8
#define LL   12
#define EE   768
#define HH   12
#define DD   64
#define FFD  3072
#define VV   50257
#define S1c  1024
#define SSc  1023

typedef __attribute__((ext_vector_type(2))) float v2f;
typedef __attribute__((ext_vector_type(8))) float v8f;

// ---------------------------------------------------------------------------
// fp32 WMMA: D(16x16,f32) = A(16x4,f32) * B(4x16,f32) + C
// A frag: lanes 0-15 hold M=lane, K=2*half..2*half+1 -> 2 floats/lane
// B frag: lanes hold N=lane%16, K=2*half..2*half+1   -> 2 floats/lane
// C frag: vgpr j, lane: M = j + 8*(lane/16), N = lane%16
// ---------------------------------------------------------------------------
__device__ inline v8f wmma4(v2f a, v2f b, v8f c) {
  return __builtin_amdgcn_wmma_f32_16x16x4_f32(
      /*neg_a=*/false, a, /*neg_b=*/false, b,
      /*c_mod=*/(short)0, c, /*reuse_a=*/false, /*reuse_b=*/false);
}

// A: [16,K] row-major (rows >= BB are zero). W: [K,N] row-major.
__device__ inline v8f tile_accum(const float* __restrict__ A,
                                 const float* __restrict__ W,
                                 int K, int N, int n0) {
  int lane = threadIdx.x & 31;
  int hl   = lane >> 4;     // 0 or 1
  int mr   = lane & 15;     // A row / B col within tile
  v8f acc = {};
  for (int k = 0; k < K; k += 4) {
    int ka = k + 2 * hl;
    v2f a, b;
    a.x = A[mr * K + ka];
    a.y = A[mr * K + ka + 1];
    const float* wp = W + (size_t)ka * N + n0 + mr;
    b.x = wp[0];
    b.y = wp[N];
    acc = wmma4(a, b, acc);
  }
  return acc;
}

// ---------------------------------------------------------------------------
// init: compute last position lp[b], embed token; pad-rows of ctx zeroed
// ---------------------------------------------------------------------------
__global__ void init_kernel(const int* __restrict__ ids, const int* __restrict__ mask,
                            const float* __restrict__ wte, const float* __restrict__ wpe,
                            int* __restrict__ lp, float* __restrict__ states,
                            float* __restrict__ ctx) {
  __shared__ int red[256];
  int b = blockIdx.x, t = threadIdx.x;
  if (b >= BB) {  // blocks 8..15: zero ctx pad rows (WMMA A-matrix padding)
    for (int e = t; e < EE; e += 256) ctx[b * EE + e] = 0.f;
    return;
  }
  int local = -1;
  for (int s = t; s < S1c; s += 256)
    if (mask[b * S1c + s] != 0 && s > local) local = s;
  red[t] = local; __syncthreads();
  for (int off = 128; off > 0; off >>= 1) {
    if (t < off) red[t] = max(red[t], red[t + off]);
    __syncthreads();
  }
  int p = red[0];
  if (t == 0) lp[b] = p;
  int tok = ids[b * S1c + p];
  for (int e = t; e < EE; e += 256)
    states[b * EE + e] = wte[(size_t)tok * EE + e] + wpe[(size_t)p * EE + e];
}

// ---------------------------------------------------------------------------
// copy kv_cache [L,2,B,S,E] -> new_kv [L,2,B,S1,E], row S1-1 zero-filled
// ---------------------------------------------------------------------------
__global__ void copykv_kernel(const float4* __restrict__ src, float4* __restrict__ dst) {
  const size_t E4 = EE / 4;
  const size_t total = (size_t)LL * 2 * BB * S1c * E4;
  size_t stride = (size_t)gridDim.x * blockDim.x;
  for (size_t i = (size_t)blockIdx.x * blockDim.x + threadIdx.x; i < total; i += stride) {
    size_t e = i % E4;
    size_t s = (i / E4) % S1c;
    size_t r = i / (E4 * S1c);           // (l*2+kv)*B + b
    float4 v;
    if (s < SSc) v = src[(r * SSc + s) * E4 + e];
    else { v.x = 0.f; v.y = 0.f; v.z = 0.f; v.w = 0.f; }
    dst[i] = v;
  }
}

// ---------------------------------------------------------------------------
// layernorm rows 0..7 of x -> out[16,E]; rows 8..15 zeroed (WMMA padding)
// ---------------------------------------------------------------------------
__global__ void ln_kernel(const float* __restrict__ x, const float* __restrict__ w,
                          const float* __restrict__ bi, float* __restrict__ out) {
  __shared__ float s1[256], s2[256];
  int r = blockIdx.x, t = threadIdx.x;
  if (r >= BB) {
    for (int e = t; e < EE; e += 256) out[r * EE + e] = 0.f;
    return;
  }
  float a = 0.f, q = 0.f;
  for (int e = t; e < EE; e += 256) { float v = x[r * EE + e]; a += v; q += v * v; }
  s1[t] = a; s2[t] = q; __syncthreads();
  for (int off = 128; off > 0; off >>= 1) {
    if (t < off) { s1[t] += s1[t + off]; s2[t] += s2[t + off]; }
    __syncthreads();
  }
  float m  = s1[0] / (float)EE;
  float vr = s2[0] / (float)EE - m * m;
  float rs = rsqrtf(vr + 1e-5f);
  for (int e = t; e < EE; e += 256)
    out[r * EE + e] = (x[r * EE + e] - m) * rs * w[e] + bi[e];
}

// ---------------------------------------------------------------------------
// QKV GEMM [16,768] x [768,2304]; q -> qbuf, k/v scattered into new_kv at lp
// one wave per 16-col tile; grid 144 x 32
// ---------------------------------------------------------------------------
__global__ void qkv_kernel(const float* __restrict__ xn, const float* __restrict__ W,
                           const float* __restrict__ bias, float* __restrict__ qbuf,
                           float* __restrict__ newkv, const int* __restrict__ lp, int l) {
  int n0 = blockIdx.x * 16;
  v8f acc = tile_accum(xn, W, EE, 3 * EE, n0);
  int lane = threadIdx.x & 31, hl = lane >> 4, mr = lane & 15;
  if (hl) return;                         // rows 8..15 are padding
  int col = n0 + mr;
  float bv = bias[col];
  for (int j = 0; j < 8; j++) {           // row j = batch b
    float val = acc[j] + bv;
    if (col < EE) {
      qbuf[j * EE + col] = val;
    } else if (col < 2 * EE) {
      int c = col - EE;
      newkv[(((size_t)l * 2 + 0) * BB + j) * (size_t)S1c * EE + (size_t)lp[j] * EE + c] = val;
    } else {
      int c = col - 2 * EE;
      newkv[(((size_t)l * 2 + 1) * BB + j) * (size_t)S1c * EE + (size_t)lp[j] * EE + c] = val;
    }
  }
}

// ---------------------------------------------------------------------------
// attention: one block per (b,h); softmax over S1 positions
// K dot uses float4 (b128) streaming loads + prefetch of the next K row
// ---------------------------------------------------------------------------
__global__ void attn_kernel(const float* __restrict__ qbuf, const float* __restrict__ newkv,
                            const int* __restrict__ mask, const float* __restrict__ ab,
                            float* __restrict__ ctx, int l) {
  __shared__ float  sc[S1c];
  __shared__ float4 qsh[DD / 4];
  __shared__ float  red[256];
  __shared__ float  s_max, s_sum;
  int b = blockIdx.x, h = blockIdx.y, t = threadIdx.x;
  const float* kbase = newkv + (((size_t)l * 2 + 0) * BB + b) * (size_t)S1c * EE + h * DD;
  const float* vbase = newkv + (((size_t)l * 2 + 1) * BB + b) * (size_t)S1c * EE + h * DD;
  if (t < DD / 4) {
    const float* qp = qbuf + b * EE + h * DD + t * 4;
    qsh[t] = make_float4(qp[0], qp[1], qp[2], qp[3]);
  }
  __syncthreads();
  float lmax = -1e30f;
  for (int s = t; s < S1c; s += 256) {
    float sv;
    if (mask[b * S1c + s] == 0) {
      sv = -1e9f;                          // reference: where(neg, -1e9, scores)
    } else {
      const float4* kp = (const float4*)(kbase + (size_t)s * EE);
      if (s + 256 < S1c)                   // gfx1250 global_prefetch of next K row
        __builtin_prefetch(kbase + (size_t)(s + 256) * EE, 0, 0);
      float a = 0.f;
      for (int d = 0; d < DD / 4; d++) {
        float4 kq = kp[d];
        float4 qq = qsh[d];
        a += qq.x * kq.x + qq.y * kq.y + qq.z * kq.z + qq.w * kq.w;
      }
      sv = a * 0.125f + ab[s];             // scale = 1/sqrt(64)
    }
    sc[s] = sv;
    if (sv > lmax) lmax = sv;
  }
  red[t] = lmax; __syncthreads();
  for (int off = 128; off > 0; off >>= 1) {
    if (t < off) red[t] = fmaxf(red[t], red[t + off]);
    __syncthreads();
  }
  if (t == 0) s_max = red[0];
  __syncthreads();
  float m = s_max, ls = 0.f;
  for (int s = t; s < S1c; s += 256) { float e = expf(sc[s] - m); sc[s] = e; ls += e; }
  red[t] = ls; __syncthreads();
  for (int off = 128; off > 0; off >>= 1) {
    if (t < off) red[t] += red[t + off];
    __syncthreads();
  }
  if (t == 0) s_sum = red[0];
  __syncthreads();
  float inv = 1.0f / s_sum;
  int d = t & (DD - 1), chunk = t >> 6;   // 4 chunks of 256 positions
  float part = 0.f;
  for (int s = chunk * 256; s < (chunk + 1) * 256; s++)
    part += sc[s] * vbase[(size_t)s * EE + d];
  red[t] = part; __syncthreads();
  if (t < DD)
    ctx[b * EE + h * DD + t] = (red[t] + red[t + 64] + red[t + 128] + red[t + 192]) * inv;
}

// ---------------------------------------------------------------------------
// residual GEMM: states[0..7,:] += A[16,K] x W[K,768] + bias ; grid 48 x 32
// ---------------------------------------------------------------------------
__global__ void gemm_res_kernel(const float* __restrict__ A, const float* __restrict__ W,
                                const float* __restrict__ bias, float* __restrict__ states,
                                int K) {
  int n0 = blockIdx.x * 16;
  v8f acc = tile_accum(A, W, K, EE, n0);
  int lane = threadIdx.x & 31, hl = lane >> 4, mr = lane & 15;
  if (hl) return;
  int col = n0 + mr;
  float bv = bias[col];
  for (int j = 0; j < 8; j++) states[j * EE + col] += acc[j] + bv;
}

// ---------------------------------------------------------------------------
// FC GEMM + exact-erf GELU -> hbuf[16,3072] (pad rows zeroed); grid 192 x 32
// ---------------------------------------------------------------------------
__global__ void fc_kernel(const float* __restrict__ A, const float* __restrict__ W,
                          const float* __restrict__ bias, float* __restrict__ out) {
  int n0 = blockIdx.x * 16;
  v8f acc = tile_accum(A, W, EE, FFD, n0);
  int lane = threadIdx.x & 31, hl = lane >> 4, mr = lane & 15;
  int col = n0 + mr;
  float bv = bias[col];
  for (int j = 0; j < 8; j++) {
    int row = j + 8 * hl;
    float x = acc[j] + bv;
    float g = 0.5f * x * (1.0f + erff(x * 0.70710678118654752f));
    out[row * FFD + col] = hl ? 0.f : g;   // keep pad rows exactly zero
  }
}

// ---------------------------------------------------------------------------
// lm_head: logits[8,V] = A[16,768] x wte^T ; wte is [V,768]; grid 3142 x 32
// ---------------------------------------------------------------------------
__global__ void lmhead_kernel(const float* __restrict__ A, const float* __restrict__ wte,
                              float* __restrict__ logits) {
  int n0 = blockIdx.x * 16;
  int lane = threadIdx.x & 31, hl = lane >> 4, mr = lane & 15;
  int col = n0 + mr;
  const float* wr = wte + (size_t)(col < VV ? col : 0) * EE;  // clamp, store-guard below
  v8f acc = {};
  for (int k = 0; k < EE; k += 4) {
    int ka = k + 2 * hl;
    v2f a, b;
    a.x = A[mr * EE + ka];
    a.y = A[mr * EE + ka + 1];
    b.x = wr[ka];
    b.y = wr[ka + 1];
    acc = wmma4(a, b, acc);
  }
  if (hl || col >= VV) return;
  for (int j = 0; j < 8; j++) logits[(size_t)j * VV + col] = acc[j];
}

// ---------------------------------------------------------------------------
// softmax + first-index argmax over V per batch row
// ---------------------------------------------------------------------------
__global__ void softmax_kernel(const float* __restrict__ logits, float* __restrict__ probs,
                               float* __restrict__ chosen) {
  __shared__ float rv[256];
  __shared__ int   ri[256];
  __shared__ float s_max, s_sum;
  __shared__ int   s_idx;
  int b = blockIdx.x, t = threadIdx.x;
  const float* row = logits + (size_t)b * VV;
  float bv = -1e30f; int bix = 0;
  for (int i = t; i < VV; i += 256) { float v = row[i]; if (v > bv) { bv = v; bix = i; } }
  rv[t] = bv; ri[t] = bix; __syncthreads();
  for (int off = 128; off > 0; off >>= 1) {
    if (t < off) {
      if (rv[t + off] > rv[t] || (rv[t + off] == rv[t] && ri[t + off] < ri[t])) {
        rv[t] = rv[t + off]; ri[t] = ri[t + off];
      }
    }
    __syncthreads();
  }
  if (t == 0) { s_max = rv[0]; s_idx = ri[0]; }
  __syncthreads();
  float m = s_max, ls = 0.f;
  for (int i = t; i < VV; i += 256) ls += expf(row[i] - m);
  rv[t] = ls; __syncthreads();
  for (int off = 128; off > 0; off >>= 1) {
    if (t < off) rv[t] += rv[t + off];
    __syncthreads();
  }
  if (t == 0) s_sum = rv[0];
  __syncthreads();
  float inv = 1.0f / s_sum;
  for (int i = t; i < VV; i += 256) probs[(size_t)b * VV + i] = expf(row[i] - m) * inv;
  if (t == 0) chosen[b] = (float)s_idx;
}

// ---------------------------------------------------------------------------
extern "C" void kernel_launch(void* const* d_in, const int* in_sizes, int n_in,
                              void* d_out, int out_size, void* d_ws, size_t ws_size,
                              hipStream_t stream) {
  const int*   ids   = (const int*)d_in[0];
  const int*   amask = (const int*)d_in[1];
  const float* kv    = (const float*)d_in[2];
  const float* wte   = (const float*)d_in[3];
  const float* wpe   = (const float*)d_in[4];
  const float* ln1w  = (const float*)d_in[5];
  const float* ln1b  = (const float*)d_in[6];
  const float* caw   = (const float*)d_in[7];
  const float* cab   = (const float*)d_in[8];
  const float* abias = (const float*)d_in[9];
  const float* cpw   = (const float*)d_in[10];
  const float* cpb   = (const float*)d_in[11];
  const float* ln2w  = (const float*)d_in[12];
  const float* ln2b  = (const float*)d_in[13];
  const float* fcw   = (const float*)d_in[14];
  const float* fcb   = (const float*)d_in[15];
  const float* prw   = (const float*)d_in[16];
  const float* prb   = (const float*)d_in[17];
  const float* lnfw  = (const float*)d_in[18];
  const float* lnfb  = (const float*)d_in[19];

  // outputs: chosen[B] (as float), probs[B,V], new_kv[L,2,B,S1,E]
  float* chosen = (float*)d_out;
  float* probs  = chosen + BB;
  float* newkv  = probs + (size_t)BB * VV;

  // workspace layout (floats)
  float* wsf    = (float*)d_ws;
  int*   lp     = (int*)wsf;             // 16 slots
  float* states = wsf + 16;              // 8  x 768
  float* xn     = states + BB * EE;      // 16 x 768
  float* qbuf   = xn + 16 * EE;          // 8  x 768
  float* ctx    = qbuf + BB * EE;        // 16 x 768
  float* xn2    = ctx + 16 * EE;         // 16 x 768
  float* hbuf   = xn2 + 16 * EE;         // 16 x 3072
  float* xnf    = hbuf + 16 * FFD;       // 16 x 768
  float* logits = xnf + 16 * EE;         // 8  x V

  init_kernel<<<16, 256, 0, stream>>>(ids, amask, wte, wpe, lp, states, ctx);
  copykv_kernel<<<2048, 256, 0, stream>>>((const float4*)kv, (float4*)newkv);

  for (int l = 0; l < LL; l++) {
    ln_kernel<<<16, 256, 0, stream>>>(states, ln1w + l * EE, ln1b + l * EE, xn);
    qkv_kernel<<<(3 * EE) / 16, 32, 0, stream>>>(
        xn, caw + (size_t)l * EE * 3 * EE, cab + (size_t)l * 3 * EE, qbuf, newkv, lp, l);
    attn_kernel<<<dim3(BB, HH), 256, 0, stream>>>(
        qbuf, newkv, amask, abias + (size_t)l * S1c, ctx, l);
    gemm_res_kernel<<<EE / 16, 32, 0, stream>>>(
        ctx, cpw + (size_t)l * EE * EE, cpb + (size_t)l * EE, states, EE);
    ln_kernel<<<16, 256, 0, stream>>>(states, ln2w + l * EE, ln2b + l * EE, xn2);
    fc_kernel<<<FFD / 16, 32, 0, stream>>>(
        xn2, fcw + (size_t)l * EE * FFD, fcb + (size_t)l * FFD, hbuf);
    gemm_res_kernel<<<EE / 16, 32, 0, stream>>>(
        hbuf, prw + (size_t)l * FFD * EE, prb + (size_t)l * EE, states, FFD);
  }

  ln_kernel<<<16, 256, 0, stream>>>(states, lnfw, lnfb, xnf);
  lmhead_kernel<<<(VV + 15) / 16, 32, 0, stream>>>(xnf, wte, logits);
  softmax_kernel<<<BB, 256, 0, stream>>>(logits, probs, chosen);
}